// SConvLayer_89146341196205
// MI455X (gfx1250) — compile-verified
//
#include <hip/hip_runtime.h>
#include <hip/hip_bf16.h>
#include <math.h>

// Problem sizes (fixed by the reference).
#define BATCH 8
#define LEN   4096
#define DIM   512
#define CHUNK 64                    // serial steps per scan chunk
#define K_CHUNKS (LEN / CHUNK)      // 64 chunks
#define TS 8                        // time-steps per async LDS tile
#define NT (CHUNK / TS)             // tiles per chunk

typedef float v2f __attribute__((ext_vector_type(2)));

#if defined(__has_builtin)
#  if __has_builtin(__builtin_amdgcn_global_load_async_to_lds_b32) && \
      __has_builtin(__builtin_amdgcn_s_wait_asynccnt)
#    define USE_ASYNC_LDS 1
#  endif
#endif
#ifndef USE_ASYNC_LDS
#  define USE_ASYNC_LDS 0
#endif

// phazor[d] = exp(-|z|^2) * (cos(angle z) + i sin(angle z)), z = phazor_param[d]
__device__ __forceinline__ void phazor_of(const float* __restrict__ pparam, int d,
                                          float& pr, float& pim,
                                          float& mag2, float& theta) {
    float a = pparam[2 * d + 0];
    float b = pparam[2 * d + 1];
    mag2  = a * a + b * b;
    theta = atan2f(b, a);
    float r = expf(-mag2);
    float s, c;
    sincosf(theta, &s, &c);
    pr  = r * c;
    pim = r * s;
}

// phazor^n in closed form: exp(-n*mag2) * (cos(n*theta) + i sin(n*theta))
__device__ __forceinline__ void phazor_pow(float mag2, float theta, float n,
                                           float& pr, float& pim) {
    float r = expf(-n * mag2);
    float s, c;
    sincosf(n * theta, &s, &c);
    pr  = r * c;
    pim = r * s;
}

// Kernel 1: per (b, chunk, d) compute the chunk's scan contribution
//   c = sum_{j=0..CHUNK-1} phazor^(CHUNK-1-j) * x[b, t0+j, d]
// (this is z at the chunk end assuming zero entering carry)
__global__ void __launch_bounds__(256)
sconv_carry_kernel(const float* __restrict__ x,
                   const float* __restrict__ pparam,
                   float2* __restrict__ carry) {
    int gid = blockIdx.x * blockDim.x + threadIdx.x;   // = ((b*K + k)*DIM + d)
    int d = gid % DIM;
    int k = (gid / DIM) % K_CHUNKS;
    int b = gid / (DIM * K_CHUNKS);

    float pr, pim, mag2, theta;
    phazor_of(pparam, d, pr, pim, mag2, theta);

    const float* xp = x + ((size_t)b * LEN + (size_t)k * CHUNK) * DIM + d;
    float cr = 0.f, ci = 0.f;
#pragma unroll 4
    for (int j = 0; j < CHUNK; ++j) {
        __builtin_prefetch(xp + 8 * DIM, 0, 0);        // global_prefetch_b8
        float xv = *xp;
        float nr = fmaf(pr, cr, fmaf(-pim, ci, xv));
        float ni = fmaf(pr, ci, pim * cr);
        cr = nr; ci = ni;
        xp += DIM;
    }
    carry[gid] = make_float2(cr, ci);
}

// Kernel 2: in-place exclusive scan of carries along k for each (b, d):
//   e[0] = 0 ; e[k+1] = phazor^CHUNK * e[k] + c[k] ; carry[k] <- e[k]
__global__ void __launch_bounds__(256)
sconv_scan_kernel(const float* __restrict__ pparam,
                  float2* __restrict__ carry) {
    int gid = blockIdx.x * blockDim.x + threadIdx.x;   // = b*DIM + d
    int d = gid % DIM;
    int b = gid / DIM;

    float pr, pim, mag2, theta;
    phazor_of(pparam, d, pr, pim, mag2, theta);
    float pcr, pci;
    phazor_pow(mag2, theta, (float)CHUNK, pcr, pci);

    float er = 0.f, ei = 0.f;
    float2* cp = carry + (size_t)b * K_CHUNKS * DIM + d;
    for (int k = 0; k < K_CHUNKS; ++k) {
        float2 t = *cp;
        *cp = make_float2(er, ei);
        float nr = fmaf(pcr, er, fmaf(-pci, ei, t.x));
        float ni = fmaf(pcr, ei, fmaf(pci, er, t.y));
        er = nr; ei = ni;
        cp += DIM;
    }
}

// Kernel 3: one workgroup per (b, chunk); thread d runs the 64-step recurrence
// seeded with its entering carry. x is streamed through LDS with async
// double-buffering (each thread loads and consumes its own LDS column, so no
// barriers are required — only s_wait_asynccnt). Outputs use non-temporal
// stores so the 192 MB of write-once output does not evict x from L2.
__global__ void __launch_bounds__(DIM)
sconv_out_kernel(const float* __restrict__ x,
                 const float* __restrict__ hidr,
                 const float* __restrict__ hidi,
                 const float* __restrict__ pinit,
                 const float* __restrict__ pparam,
                 const float2* __restrict__ carry,
                 float* __restrict__ out) {
    extern __shared__ float tile[];                    // 2 * TS * DIM floats
    int d = threadIdx.x;
    int k = blockIdx.x % K_CHUNKS;
    int b = blockIdx.x / K_CHUNKS;
    int t0 = k * CHUNK;

    float pr, pim, mag2, theta;
    phazor_of(pparam, d, pr, pim, mag2, theta);
    float fir = pinit[2 * d + 0];
    float fii = pinit[2 * d + 1];
    float hr = hidr[b * DIM + d];
    float hi = hidi[b * DIM + d];

    // q = hidden * phazor^(t0+1); advanced by *phazor each step
    float p0r, p0i;
    phazor_pow(mag2, theta, (float)(t0 + 1), p0r, p0i);
    float qr = hr * p0r - hi * p0i;
    float qi = hr * p0i + hi * p0r;

    float2 e = carry[((size_t)b * K_CHUNKS + k) * DIM + d];
    float zr = e.x, zi = e.y;

    const float* xp = x + ((size_t)b * LEN + t0) * DIM + d;
    float* outR = out;                                        // [B,L,D] real
    float* outC = out + (size_t)BATCH * LEN * DIM;            // [B,L,D] (re,im)
    size_t idx0 = ((size_t)b * LEN + t0) * DIM + d;

#if USE_ASYNC_LDS
    auto issue_tile = [&](int tt, int buf) {
        const float* g = xp + (size_t)tt * TS * DIM;
        float* s = tile + buf * (TS * DIM) + d;
#pragma unroll
        for (int i = 0; i < TS; ++i) {
            __builtin_amdgcn_global_load_async_to_lds_b32(
                (__attribute__((address_space(1))) int*)(g + (size_t)i * DIM),
                (__attribute__((address_space(3))) int*)(s + i * DIM),
                0, 0);
        }
    };
    issue_tile(0, 0);
    for (int tt = 0; tt < NT; ++tt) {
        int buf = tt & 1;
        if (tt + 1 < NT) {
            issue_tile(tt + 1, (tt + 1) & 1);
            __builtin_amdgcn_s_wait_asynccnt(TS);   // current tile landed
        } else {
            __builtin_amdgcn_s_wait_asynccnt(0);
        }
        const float* sp = tile + buf * (TS * DIM) + d;
#pragma unroll
        for (int i = 0; i < TS; ++i) {
            float xv = sp[i * DIM];
            float nzr = fmaf(pr, zr, fmaf(-pim, zi, xv));
            float nzi = fmaf(pr, zi, pim * zr);
            zr = nzr; zi = nzi;
            float orr = fmaf(fir, zr, fmaf(-fii, zi, qr));
            float oii = fmaf(fir, zi, fmaf(fii, zr, qi));
            size_t idx = idx0 + (size_t)(tt * TS + i) * DIM;
            __builtin_nontemporal_store(orr, outR + idx);
            v2f oc;
            oc.x = orr; oc.y = oii;
            __builtin_nontemporal_store(oc, (v2f*)(outC + 2 * idx));
            float nqr = pr * qr - pim * qi;
            float nqi = pr * qi + pim * qr;
            qr = nqr; qi = nqi;
        }
    }
#else
#pragma unroll 4
    for (int j = 0; j < CHUNK; ++j) {
        __builtin_prefetch(xp + (size_t)(j + 8) * DIM, 0, 0);
        float xv = xp[(size_t)j * DIM];
        float nzr = fmaf(pr, zr, fmaf(-pim, zi, xv));
        float nzi = fmaf(pr, zi, pim * zr);
        zr = nzr; zi = nzi;
        float orr = fmaf(fir, zr, fmaf(-fii, zi, qr));
        float oii = fmaf(fir, zi, fmaf(fii, zr, qi));
        size_t idx = idx0 + (size_t)j * DIM;
        __builtin_nontemporal_store(orr, outR + idx);
        v2f oc;
        oc.x = orr; oc.y = oii;
        __builtin_nontemporal_store(oc, (v2f*)(outC + 2 * idx));
        float nqr = pr * qr - pim * qi;
        float nqi = pr * qi + pim * qr;
        qr = nqr; qi = nqi;
    }
#endif
}

extern "C" void kernel_launch(void* const* d_in, const int* in_sizes, int n_in,
                              void* d_out, int out_size, void* d_ws, size_t ws_size,
                              hipStream_t stream) {
    const float* x      = (const float*)d_in[0];   // [B,L,D]
    const float* hr     = (const float*)d_in[1];   // [B,D]
    const float* hi     = (const float*)d_in[2];   // [B,D]
    const float* pinit  = (const float*)d_in[3];   // [D,2]
    const float* pparam = (const float*)d_in[4];   // [D,2]
    float* out = (float*)d_out;                    // B*L*D real + B*L*D*2 complex
    float2* carry = (float2*)d_ws;                 // [B, K_CHUNKS, D] complex (2 MB)

    sconv_carry_kernel<<<(BATCH * K_CHUNKS * DIM) / 256, 256, 0, stream>>>(
        x, pparam, carry);
    sconv_scan_kernel<<<(BATCH * DIM) / 256, 256, 0, stream>>>(pparam, carry);
    sconv_out_kernel<<<BATCH * K_CHUNKS, DIM, 2 * TS * DIM * sizeof(float), stream>>>(
        x, hr, hi, pinit, pparam, carry, out);
}